// RWKV6TimeMix_75496935129738
// MI455X (gfx1250) — compile-verified
//
#include <hip/hip_runtime.h>
#include <stdint.h>

typedef __attribute__((ext_vector_type(2))) float v2f;
typedef __attribute__((ext_vector_type(8))) float v8f;
typedef __attribute__((ext_vector_type(4))) float f4;

#define Bn 4
#define Hn 16
#define Tn 2048
#define Kn 64
#define TC 128
#define NCH (Tn / TC)
#define NT 512
#define STR 68    // padded LDS row stride (floats) for 64-col tiles: 16B-aligned rows, 4-bank rotation
#define STRA 132  // padded LDS row stride (floats) for the 128-col A matrix
// log(0.005)
#define LOG_DECAY_MIN (-5.2983173665480363f)

// One workgroup per (b,h) stream; sequential scan over NCH chunks.
// Matmuls on V_WMMA_F32_16X16X4_F32 (fp32 fidelity vs fp32 reference);
// input staging via GLOBAL_LOAD_ASYNC_TO_LDS_B128 (ASYNCcnt path).
__global__ __launch_bounds__(NT) void rwkv6_chunked_wkv(
    const float* __restrict__ r, const float* __restrict__ k,
    const float* __restrict__ v, const float* __restrict__ w,
    const float* __restrict__ u, const float* __restrict__ s0,
    float* __restrict__ out, float* __restrict__ fstate)
{
    // ~233 KB of LDS -- fits in CDNA5's 320 KB/WGP
    __shared__ __align__(16) float sR[TC * STR];     // rq = r * exp(shifted - off)
    __shared__ __align__(16) float sK[TC * STR];     // kq = k * exp(off - cum)
    __shared__ __align__(16) float sV[TC * STR];     // raw v
    __shared__ __align__(16) float sA[TC * STRA];    // masked attention matrix
    __shared__ __align__(16) float sCum[TC * Kn];    // raw w, overwritten in place by cumsum
    __shared__ __align__(16) float sState[Kn * Kn];  // running wkv state
    __shared__ __align__(16) float sWKV[Kn * Kn];    // per-chunk K^T V
    __shared__ float sU[Kn];
    __shared__ float sDiag[TC];
    __shared__ float sEOff[Kn];    // exp(off)      : rw = rq * sEOff
    __shared__ float sEWsOff[Kn];  // exp(ws - off) : kw = kq * sEWsOff
    __shared__ float sEWs[Kn];     // exp(ws)       : state decay

    const int tid  = threadIdx.x;
    const int bh   = blockIdx.x;
    const int h    = bh % Hn;
    const int wave = tid >> 5;
    const int lane = tid & 31;
    const int lh   = lane & 15;
    const int hiK  = (lane >= 16) ? 2 : 0;  // K-column offset for A/B fragments
    const int hiM  = (lane >= 16) ? 8 : 0;  // M-row offset for C/D fragments

    const size_t base = (size_t)bh * Tn * Kn;

    {
        const f4* s4 = (const f4*)(s0 + (size_t)bh * Kn * Kn);
        for (int i = tid; i < (Kn * Kn) / 4; i += NT) ((f4*)sState)[i] = s4[i];
    }
    if (tid < Kn) sU[tid] = u[h * Kn + tid];
    __syncthreads();

    const uint32_t ldsR = (uint32_t)(uintptr_t)sR;
    const uint32_t ldsK = (uint32_t)(uintptr_t)sK;
    const uint32_t ldsV = (uint32_t)(uintptr_t)sV;
    const uint32_t ldsW = (uint32_t)(uintptr_t)sCum;

    for (int n = 0; n < NCH; ++n) {
        const size_t coff = base + (size_t)n * TC * Kn;

        // 1. stage chunk inputs straight into LDS via async B128 copies
        //    (each lane moves one float4 per array per pass; 4 passes)
        for (int i = tid; i < (TC * Kn) / 4; i += NT) {
            const int      row  = i >> 4;          // 16 float4 per 64-wide row
            const int      c4   = (i & 15) << 2;   // column of this float4
            const uint32_t dRKV = (uint32_t)((row * STR + c4) * 4);  // padded dest (bytes)
            const uint32_t gb   = (uint32_t)(i << 4);                // source offset (bytes)
            const uint32_t aR = ldsR + dRKV;
            const uint32_t aK = ldsK + dRKV;
            const uint32_t aV = ldsV + dRKV;
            const uint32_t aW = ldsW + gb;  // w is staged dense into sCum
            asm volatile("global_load_async_to_lds_b128 %0, %1, %2"
                         :: "v"(aR), "v"(gb), "s"(r + coff) : "memory");
            asm volatile("global_load_async_to_lds_b128 %0, %1, %2"
                         :: "v"(aK), "v"(gb), "s"(k + coff) : "memory");
            asm volatile("global_load_async_to_lds_b128 %0, %1, %2"
                         :: "v"(aV), "v"(gb), "s"(v + coff) : "memory");
            asm volatile("global_load_async_to_lds_b128 %0, %1, %2"
                         :: "v"(aW), "v"(gb), "s"(w + coff) : "memory");
        }
        asm volatile("s_wait_asynccnt 0x0" ::: "memory");
        __syncthreads();

        // 2. per-feature cumsum of clipped log-decay, in place
        //    (log(clip(exp(w),.005)) == max(w, log .005))
        if (tid < Kn) {
            float c = 0.f;
            for (int t = 0; t < TC; ++t) {
                c += fmaxf(sCum[t * Kn + tid], LOG_DECAY_MIN);
                sCum[t * Kn + tid] = c;
            }
            const float wsum = c;
            const float off  = sCum[(TC / 2 - 1) * Kn + tid];  // shifted[Tc/2]
            sEOff[tid]   = expf(off);
            sEWs[tid]    = expf(wsum);
            sEWsOff[tid] = expf(wsum - off);
        }
        __syncthreads();

        // 3. diagonal terms from raw r,k: diag[t] = sum_k r*u*k
        if (tid < TC) {
            float d = 0.f;
            for (int kk = 0; kk < Kn; ++kk)
                d += sR[tid * STR + kk] * sU[kk] * sK[tid * STR + kk];
            sDiag[tid] = d;
        }
        __syncthreads();

        // 4. fold decay factors into staged r,k
        for (int e = tid; e < TC * Kn; e += NT) {
            const int   t   = e >> 6;
            const int   kf  = e & (Kn - 1);
            const float cum = sCum[e];
            const float sh  = (t == 0) ? 0.f : sCum[e - Kn];
            const float off = sCum[(TC / 2 - 1) * Kn + kf];
            sR[t * STR + kf] *= expf(sh - off);
            sK[t * STR + kf] *= expf(off - cum);
        }
        __syncthreads();

        // 5. A = tril(rq @ kq^T, -1) + diag  -- only lower-triangular tiles computed
        {
            const int ti = wave >> 1;
            for (int tj = (wave & 1); tj <= ti; tj += 2) {
                v8f acc = {};
                for (int kk = 0; kk < Kn / 4; ++kk) {
                    const int ka = kk * 4 + hiK;
                    v2f a; a.x = sR[(ti * 16 + lh) * STR + ka];
                           a.y = sR[(ti * 16 + lh) * STR + ka + 1];
                    v2f b; b.x = sK[(tj * 16 + lh) * STR + ka];
                           b.y = sK[(tj * 16 + lh) * STR + ka + 1];
                    acc = __builtin_amdgcn_wmma_f32_16x16x4_f32(
                        false, a, false, b, (short)0, acc, false, false);
                }
                const int ncol = tj * 16 + lh;
                for (int vg = 0; vg < 8; ++vg) {
                    const int mrow = ti * 16 + vg + hiM;
                    float val = acc[vg];
                    if (mrow == ncol)      val = sDiag[mrow];
                    else if (mrow < ncol)  val = 0.f;
                    sA[mrow * STRA + ncol] = val;
                }
            }
        }
        __syncthreads();

        // 6. out = A @ V  +  (rq * expOff) @ state_prev   (store straight to HBM)
        {
            const int ti = wave >> 1;
            for (int it = 0; it < 2; ++it) {
                const int tc = (wave & 1) + 2 * it;
                v8f acc = {};
                // intra-chunk: only time tiles jt <= ti contribute (tril)
                for (int jt = 0; jt <= ti; ++jt) {
                    for (int s = 0; s < 4; ++s) {
                        const int kt = jt * 16 + s * 4 + hiK;
                        v2f a; a.x = sA[(ti * 16 + lh) * STRA + kt];
                               a.y = sA[(ti * 16 + lh) * STRA + kt + 1];
                        v2f b; b.x = sV[kt * STR + tc * 16 + lh];
                               b.y = sV[(kt + 1) * STR + tc * 16 + lh];
                        acc = __builtin_amdgcn_wmma_f32_16x16x4_f32(
                            false, a, false, b, (short)0, acc, false, false);
                    }
                }
                // inter-chunk: rw @ state_prev (rw = rq * exp(off) per feature)
                for (int kk = 0; kk < Kn / 4; ++kk) {
                    const int kf = kk * 4 + hiK;
                    v2f a; a.x = sR[(ti * 16 + lh) * STR + kf]     * sEOff[kf];
                           a.y = sR[(ti * 16 + lh) * STR + kf + 1] * sEOff[kf + 1];
                    v2f b; b.x = sState[kf * Kn + tc * 16 + lh];
                           b.y = sState[(kf + 1) * Kn + tc * 16 + lh];
                    acc = __builtin_amdgcn_wmma_f32_16x16x4_f32(
                        false, a, false, b, (short)0, acc, false, false);
                }
                for (int vg = 0; vg < 8; ++vg) {
                    const int mrow = ti * 16 + vg + hiM;
                    out[coff + mrow * Kn + tc * 16 + lh] = acc[vg];
                }
            }
        }

        // 7. wkv = (kq * expWsOff)^T @ V  -- one 16x16 tile per wave
        {
            const int fi = wave >> 2;
            const int vj = wave & 3;
            v8f acc = {};
            const float ew = sEWsOff[fi * 16 + lh];  // per A-row (feature) scale
            for (int kk = 0; kk < TC / 4; ++kk) {
                const int tt = kk * 4 + hiK;
                v2f a; a.x = sK[tt * STR + fi * 16 + lh]       * ew;
                       a.y = sK[(tt + 1) * STR + fi * 16 + lh] * ew;
                v2f b; b.x = sV[tt * STR + vj * 16 + lh];
                       b.y = sV[(tt + 1) * STR + vj * 16 + lh];
                acc = __builtin_amdgcn_wmma_f32_16x16x4_f32(
                    false, a, false, b, (short)0, acc, false, false);
            }
            for (int vg = 0; vg < 8; ++vg) {
                const int mrow = fi * 16 + vg + hiM;
                sWKV[mrow * Kn + vj * 16 + lh] = acc[vg];
            }
        }
        __syncthreads();  // all reads of state_prev (step 6) done; sWKV visible

        // 8. state = state * exp(ws) + wkv
        for (int e = tid; e < Kn * Kn; e += NT) {
            const int kf = e >> 6;
            sState[e] = sState[e] * sEWs[kf] + sWKV[e];
        }
        __syncthreads();
    }

    {
        f4* f4out = (f4*)(fstate + (size_t)bh * Kn * Kn);
        for (int i = tid; i < (Kn * Kn) / 4; i += NT) f4out[i] = ((const f4*)sState)[i];
    }
}

extern "C" void kernel_launch(void* const* d_in, const int* in_sizes, int n_in,
                              void* d_out, int out_size, void* d_ws, size_t ws_size,
                              hipStream_t stream) {
    (void)in_sizes; (void)n_in; (void)d_ws; (void)ws_size; (void)out_size;
    const float* r  = (const float*)d_in[0];
    const float* k  = (const float*)d_in[1];
    const float* v  = (const float*)d_in[2];
    const float* w  = (const float*)d_in[3];
    const float* u  = (const float*)d_in[4];
    const float* s0 = (const float*)d_in[5];

    float* out    = (float*)d_out;
    float* fstate = out + (size_t)Bn * Hn * Tn * Kn;  // tuple: (out, final_state)

    rwkv6_chunked_wkv<<<dim3(Bn * Hn), dim3(NT), 0, stream>>>(
        r, k, v, w, u, s0, out, fstate);
}